// SLayer_45200235823404
// MI455X (gfx1250) — compile-verified
//
#include <hip/hip_runtime.h>

// Problem constants from the reference: B=64, P=16384, N=64, D=2
#define BATCHES 64
#define NPOINTS 16384
#define NCENT   64

#define PCHUNKS          8   // gridDim.y
#define WAVES_PER_BLOCK  8   // blockDim = 256
#define TILE_STRIDE      (PCHUNKS * WAVES_PER_BLOCK)                  // 64
#define TILES_PER_WAVE   ((NPOINTS / 16) / TILE_STRIDE)               // 16
#define L2E              1.4426950408889634f

typedef __attribute__((ext_vector_type(2))) float v2f;
typedef __attribute__((ext_vector_type(8))) float v8f;

__global__ void slayer_zero_out(float* __restrict__ out) {
    int i = blockIdx.x * blockDim.x + threadIdx.x;
    if (i < BATCHES * NCENT) out[i] = 0.0f;
}

// V_WMMA_F32_16X16X4_F32 computes (in log2 domain, centers on M, points on N):
//   D[m][n] = L2E * ( -(s0*c0^2+s1*c1^2) + 2s0c0*x0 + 2s1c1*x1 - s0*x0^2 - s1*x1^2 )
// then w = exp2(D) * mask, summed over points. A (coeffs) and C (constants) are
// loop-invariant registers; B carries the per-tile point features.
__global__ __launch_bounds__(256)
void slayer_wmma_kernel(const float* __restrict__ batch,     // [B, P, 2]
                        const float* __restrict__ mask,      // [B, P]
                        const float* __restrict__ centers,   // [N, 2]
                        const float* __restrict__ sharp,     // [N, 2]
                        float* __restrict__ out) {           // [B, N]
    const int b    = blockIdx.x;
    const int lane = threadIdx.x & 31;
    const int wave = threadIdx.x >> 5;
    const int nl   = lane & 15;     // point column (N) / center id within tile for A
    const int h    = lane >> 4;     // half-wave: selects K=2,3 slots and M rows r+8

    __shared__ float red[NCENT];
    if (threadIdx.x < NCENT) red[threadIdx.x] = 0.0f;
    __syncthreads();

    // ---- Loop-invariant WMMA operands: A = center coefficients, C = constants ----
    v2f amat[4];
    v8f cmat[4];
    v8f acc[4];
#pragma unroll
    for (int t = 0; t < 4; ++t) {
        {   // A layout: lane holds M = lane%16; VGPR0/1 hold K = 2h+0 / 2h+1
            const int n  = t * 16 + nl;
            const float cx = centers[2 * n], cy = centers[2 * n + 1];
            const float sx = sharp[2 * n],   sy = sharp[2 * n + 1];
            amat[t].x = h ? (-L2E * sx) : (2.0f * L2E * sx * cx);
            amat[t].y = h ? (-L2E * sy) : (2.0f * L2E * sy * cy);
        }
#pragma unroll
        for (int r = 0; r < 8; ++r) {
            // C/D layout: VGPR r holds M = r + 8h (same value across point columns)
            const int n  = t * 16 + r + 8 * h;
            const float cx = centers[2 * n], cy = centers[2 * n + 1];
            const float sx = sharp[2 * n],   sy = sharp[2 * n + 1];
            cmat[t][r] = -L2E * (sx * cx * cx + sy * cy * cy);
            acc[t][r]  = 0.0f;
        }
    }

    const size_t bbase = (size_t)b * NPOINTS;
    const float2* __restrict__ bptr = (const float2*)batch;

    int p = (blockIdx.y * WAVES_PER_BLOCK + wave) * 16 + nl;

    // Software pipeline: prefetch next tile's point + mask while computing current
    float2 xv = bptr[bbase + p];
    float  mk = mask[bbase + p];

    for (int i = 0; i < TILES_PER_WAVE; ++i) {
        const float x0 = xv.x, x1 = xv.y, m = mk;
        p += TILE_STRIDE * 16;
        if (i + 1 < TILES_PER_WAVE) {      // wave-uniform condition
            xv = bptr[bbase + p];
            mk = mask[bbase + p];
        }

        // B layout: lane holds N = lane%16; VGPR0/1 hold K = 2h+0 / 2h+1
        v2f bfeat;
        bfeat.x = h ? x0 * x0 : x0;
        bfeat.y = h ? x1 * x1 : x1;

#pragma unroll
        for (int t = 0; t < 4; ++t) {
            v8f d = __builtin_amdgcn_wmma_f32_16x16x4_f32(
                false, amat[t], false, bfeat, (short)0, cmat[t], false, false);
#pragma unroll
            for (int r = 0; r < 8; ++r)
                acc[t][r] += __builtin_amdgcn_exp2f(d[r]) * m;  // bare v_exp_f32
        }
    }

    // Epilogue: butterfly-reduce over the 16 point columns inside each half-wave
#pragma unroll
    for (int t = 0; t < 4; ++t) {
#pragma unroll
        for (int r = 0; r < 8; ++r) {
            float v = acc[t][r];
            v += __shfl_xor(v, 1, 32);
            v += __shfl_xor(v, 2, 32);
            v += __shfl_xor(v, 4, 32);
            v += __shfl_xor(v, 8, 32);
            acc[t][r] = v;
        }
    }

    if (nl == 0) {   // lanes 0 and 16 carry the two M-halves (centers r / r+8)
#pragma unroll
        for (int t = 0; t < 4; ++t)
#pragma unroll
            for (int r = 0; r < 8; ++r)
                atomicAdd(&red[t * 16 + r + 8 * h], acc[t][r]);
    }
    __syncthreads();
    if (threadIdx.x < NCENT)
        atomicAdd(&out[b * NCENT + threadIdx.x], red[threadIdx.x]);
}

extern "C" void kernel_launch(void* const* d_in, const int* in_sizes, int n_in,
                              void* d_out, int out_size, void* d_ws, size_t ws_size,
                              hipStream_t stream) {
    (void)in_sizes; (void)n_in; (void)out_size; (void)d_ws; (void)ws_size;
    const float* batch   = (const float*)d_in[0];
    const float* mask    = (const float*)d_in[1];
    const float* centers = (const float*)d_in[2];
    const float* sharp   = (const float*)d_in[3];
    float* out = (float*)d_out;

    hipLaunchKernelGGL(slayer_zero_out, dim3((BATCHES * NCENT + 255) / 256),
                       dim3(256), 0, stream, out);

    hipLaunchKernelGGL(slayer_wmma_kernel, dim3(BATCHES, PCHUNKS), dim3(256),
                       0, stream, batch, mask, centers, sharp, out);
}